// Boftrainer_48284022342309
// MI455X (gfx1250) — compile-verified
//
#include <hip/hip_runtime.h>
#include <hip/hip_bf16.h>

typedef __attribute__((ext_vector_type(2))) float v2f;
typedef __attribute__((ext_vector_type(8))) float v8f;

// ---------------------------------------------------------------------------
// 3x3 same-padded conv + ReLU. Weights/bias staged in LDS.
// f_layout=1 writes out[(b*H*W + h*W + w)*Cout + c]  (feature layout for WMMA A)
// ---------------------------------------------------------------------------
__global__ __launch_bounds__(256) void conv3x3_relu(
    const float* __restrict__ in, const float* __restrict__ wgt,
    const float* __restrict__ bias, float* __restrict__ out,
    int B, int Cin, int Cout, int H, int W, int f_layout)
{
    __shared__ float s_w[24 * 24 * 9];   // max 24*24*9 > any layer here
    __shared__ float s_b[32];
    const int nw = Cout * Cin * 9;
    for (int i = threadIdx.x; i < nw; i += blockDim.x) s_w[i] = wgt[i];
    for (int i = threadIdx.x; i < Cout; i += blockDim.x) s_b[i] = bias[i];
    __syncthreads();

    const long total = (long)B * Cout * H * W;
    const long idx = (long)blockIdx.x * blockDim.x + threadIdx.x;
    if (idx >= total) return;

    const int w = (int)(idx % W);
    const int h = (int)((idx / W) % H);
    const int c = (int)((idx / ((long)W * H)) % Cout);
    const int b = (int)(idx / ((long)W * H * Cout));

    float acc = s_b[c];
    const float* inb = in + (long)b * Cin * H * W;
    const float* wc  = s_w + c * Cin * 9;
    for (int ci = 0; ci < Cin; ++ci) {
        const float* ip = inb + (long)ci * H * W;
        const float* wp = wc + ci * 9;
        #pragma unroll
        for (int dy = -1; dy <= 1; ++dy) {
            const int y = h + dy;
            if (y < 0 || y >= H) continue;
            const float* row = ip + (long)y * W;
            #pragma unroll
            for (int dx = -1; dx <= 1; ++dx) {
                const int x = w + dx;
                if (x < 0 || x >= W) continue;
                acc = fmaf(row[x], wp[(dy + 1) * 3 + (dx + 1)], acc);
            }
        }
    }
    acc = fmaxf(acc, 0.0f);
    if (f_layout)
        out[((long)b * (H * W) + (long)h * W + w) * Cout + c] = acc;
    else
        out[idx] = acc;
}

// ---------------------------------------------------------------------------
// 2x2 max pool, stride 2. in: (BC, 2H, 2W) -> out: (BC, H, W)
// ---------------------------------------------------------------------------
__global__ __launch_bounds__(256) void maxpool2x2(
    const float* __restrict__ in, float* __restrict__ out,
    int total, int H, int W)
{
    const int idx = blockIdx.x * blockDim.x + threadIdx.x;
    if (idx >= total) return;
    const int w  = idx % W;
    const int h  = (idx / W) % H;
    const int bc = idx / (W * H);
    const float* ip = in + ((long)bc * (2 * H) + 2 * h) * (2 * W) + 2 * w;
    out[idx] = fmaxf(fmaxf(ip[0], ip[1]), fmaxf(ip[2 * W], ip[2 * W + 1]));
}

// ---------------------------------------------------------------------------
// RBF bag-of-features stage using V_WMMA_F32_16X16X4_F32.
// One block per image b. 8 waves; each wave owns 16 pixels per tile, loops
// over 64/8 = 8 pixel tiles. Per tile: pass1 accumulates sum_K exp(-d*sigma)
// (L1 norm denominator), pass2 recomputes and accumulates normalized a into
// per-lane registers, committed once via LDS float atomics.
// f: (B, 1024, 16) row-major.  pooled: (B, 512).
// ---------------------------------------------------------------------------
__global__ __launch_bounds__(256) void rbf_wmma(
    const float* __restrict__ f, const float* __restrict__ codebook,
    const float* __restrict__ sigma, float* __restrict__ pooled)
{
    __shared__ __align__(16) float s_cb[512 * 18];  // padded rows: bank-friendly
    __shared__ float s_c2[512];
    __shared__ float s_sig[512];
    __shared__ float s_pooled[512];

    const int tid = threadIdx.x;
    const int b   = blockIdx.x;

    for (int i = tid; i < 512 * 16; i += 256) {
        const int k = i >> 4, d = i & 15;
        s_cb[k * 18 + d] = codebook[i];
    }
    for (int i = tid; i < 512; i += 256) { s_sig[i] = sigma[i]; s_pooled[i] = 0.0f; }
    __syncthreads();
    for (int k = tid; k < 512; k += 256) {
        float s = 0.0f;
        #pragma unroll
        for (int d = 0; d < 16; ++d) { const float v = s_cb[k * 18 + d]; s = fmaf(v, v, s); }
        s_c2[k] = s;
    }
    __syncthreads();

    const int lane = tid & 31;
    const int half = lane >> 4;      // 0: m=0..7 of D tile, 1: m=8..15
    const int ln   = lane & 15;
    const int koff = half * 2;       // A/B operand K-offset per ISA 16x16x4 layout
    const int wave = tid >> 5;

    float aw[32];
    #pragma unroll
    for (int t = 0; t < 32; ++t) aw[t] = 0.0f;

    for (int pt = wave; pt < 64; pt += 8) {            // uniform per wave: EXEC all-1
        const int p = pt * 16 + ln;                    // pixel row m = ln of this tile
        const float* fr = f + ((size_t)b * 1024 + p) * 16;

        v2f af[4];
        float f2p = 0.0f;
        #pragma unroll
        for (int c = 0; c < 4; ++c) {
            af[c] = *(const v2f*)(fr + 4 * c + koff);
            f2p = fmaf(af[c].x, af[c].x, fmaf(af[c].y, af[c].y, f2p));
        }
        // lanes l and l+16 hold complementary K-halves of pixel row l
        const float f2full = f2p + __shfl_xor(f2p, 16);
        float f2m[8];
        #pragma unroll
        for (int r = 0; r < 8; ++r) f2m[r] = __shfl(f2full, half * 8 + r);

        // ---- pass 1: L1 denominator per pixel ----
        float ks[8];
        #pragma unroll
        for (int r = 0; r < 8; ++r) ks[r] = 0.0f;
        for (int t = 0; t < 32; ++t) {
            const int code = t * 16 + ln;
            const float* cr = s_cb + code * 18 + koff;
            v8f acc = {0.f, 0.f, 0.f, 0.f, 0.f, 0.f, 0.f, 0.f};
            #pragma unroll
            for (int c = 0; c < 4; ++c) {
                const v2f bf = *(const v2f*)(cr + 4 * c);
                acc = __builtin_amdgcn_wmma_f32_16x16x4_f32(
                        false, af[c], false, bf, (short)0, acc, false, false);
            }
            const float c2n = s_c2[code], sg = s_sig[code];
            #pragma unroll
            for (int r = 0; r < 8; ++r) {
                const float dist = f2m[r] + c2n - 2.0f * acc[r];
                ks[r] += __expf(-dist * sg);
            }
        }
        // reduce over the 16 code-lanes of this half (m-partition matches D layout)
        float rinv[8];
        #pragma unroll
        for (int r = 0; r < 8; ++r) {
            float s = ks[r];
            s += __shfl_xor(s, 1); s += __shfl_xor(s, 2);
            s += __shfl_xor(s, 4); s += __shfl_xor(s, 8);
            rinv[r] = 1.0f / fmaxf(s, 1e-12f);
        }

        // ---- pass 2: recompute, normalize, accumulate over pixels ----
        for (int t = 0; t < 32; ++t) {
            const int code = t * 16 + ln;
            const float* cr = s_cb + code * 18 + koff;
            v8f acc = {0.f, 0.f, 0.f, 0.f, 0.f, 0.f, 0.f, 0.f};
            #pragma unroll
            for (int c = 0; c < 4; ++c) {
                const v2f bf = *(const v2f*)(cr + 4 * c);
                acc = __builtin_amdgcn_wmma_f32_16x16x4_f32(
                        false, af[c], false, bf, (short)0, acc, false, false);
            }
            const float c2n = s_c2[code], sg = s_sig[code];
            float part = 0.0f;
            #pragma unroll
            for (int r = 0; r < 8; ++r) {
                const float dist = f2m[r] + c2n - 2.0f * acc[r];
                part = fmaf(__expf(-dist * sg), rinv[r], part);
            }
            part += __shfl_xor(part, 16);   // combine m=0..7 with m=8..15
            aw[t] += part;                  // both halves track it; half 0 commits
        }
    }

    if (half == 0) {
        #pragma unroll
        for (int t = 0; t < 32; ++t)
            atomicAdd(&s_pooled[t * 16 + ln], aw[t]);   // ds_add_f32
    }
    __syncthreads();
    const float invN = 1.0f / 1024.0f;
    for (int k = tid; k < 512; k += 256)
        pooled[(size_t)b * 512 + k] = s_pooled[k] * invN;
}

// ---------------------------------------------------------------------------
// Tiny FC head: relu(pooled @ l1_w^T + l1_b) @ l2_w^T + l2_b.  One block per b.
// ---------------------------------------------------------------------------
__global__ __launch_bounds__(64) void fc_head(
    const float* __restrict__ pooled,
    const float* __restrict__ l1w, const float* __restrict__ l1b,
    const float* __restrict__ l2w, const float* __restrict__ l2b,
    float* __restrict__ out)
{
    __shared__ float s_p[512];
    __shared__ float s_z[20];
    const int b = blockIdx.x, tid = threadIdx.x;
    for (int i = tid; i < 512; i += 64) s_p[i] = pooled[(size_t)b * 512 + i];
    __syncthreads();
    if (tid < 20) {
        float a = l1b[tid];
        for (int k = 0; k < 512; ++k) a = fmaf(s_p[k], l1w[tid * 512 + k], a);
        s_z[tid] = fmaxf(a, 0.0f);
    }
    __syncthreads();
    if (tid < 10) {
        float a = l2b[tid];
        #pragma unroll
        for (int j = 0; j < 20; ++j) a = fmaf(s_z[j], l2w[tid * 20 + j], a);
        out[b * 10 + tid] = a;
    }
}

// ---------------------------------------------------------------------------
extern "C" void kernel_launch(void* const* d_in, const int* in_sizes, int n_in,
                              void* d_out, int out_size, void* d_ws, size_t ws_size,
                              hipStream_t stream)
{
    const float* x    = (const float*)d_in[0];
    const float* w1   = (const float*)d_in[1];
    const float* b1   = (const float*)d_in[2];
    const float* w2   = (const float*)d_in[3];
    const float* b2   = (const float*)d_in[4];
    const float* w3   = (const float*)d_in[5];
    const float* b3   = (const float*)d_in[6];
    const float* w4   = (const float*)d_in[7];
    const float* b4   = (const float*)d_in[8];
    const float* cb   = (const float*)d_in[9];
    const float* sg   = (const float*)d_in[10];
    const float* l1w  = (const float*)d_in[11];
    const float* l1b  = (const float*)d_in[12];
    const float* l2w  = (const float*)d_in[13];
    const float* l2b  = (const float*)d_in[14];
    float* outp = (float*)d_out;

    const size_t BUF = (size_t)64 * 16 * 64 * 64 * sizeof(float);  // 16 MiB
    float* bufA   = (float*)d_ws;
    float* bufB   = (float*)((char*)d_ws + BUF);
    float* pooled = (float*)((char*)d_ws + 2 * BUF);               // 64*512 floats

    // conv1: (64,3,64,64) -> (64,16,64,64)
    {
        const int total = 64 * 16 * 64 * 64;
        conv3x3_relu<<<(total + 255) / 256, 256, 0, stream>>>(
            x, w1, b1, bufA, 64, 3, 16, 64, 64, 0);
    }
    // conv2: (64,16,64,64) -> (64,16,64,64)
    {
        const int total = 64 * 16 * 64 * 64;
        conv3x3_relu<<<(total + 255) / 256, 256, 0, stream>>>(
            bufA, w2, b2, bufB, 64, 16, 16, 64, 64, 0);
    }
    // maxpool -> (64,16,32,32)
    {
        const int total = 64 * 16 * 32 * 32;
        maxpool2x2<<<(total + 255) / 256, 256, 0, stream>>>(bufB, bufA, total, 32, 32);
    }
    // conv3: -> (64,24,32,32)
    {
        const int total = 64 * 24 * 32 * 32;
        conv3x3_relu<<<(total + 255) / 256, 256, 0, stream>>>(
            bufA, w3, b3, bufB, 64, 16, 24, 32, 32, 0);
    }
    // conv4: -> features f (64, 1024, 16)
    {
        const int total = 64 * 16 * 32 * 32;
        conv3x3_relu<<<(total + 255) / 256, 256, 0, stream>>>(
            bufB, w4, b4, bufA, 64, 24, 16, 32, 32, 1);
    }
    // RBF membership + L1 normalize + mean pool, fp32 WMMA
    rbf_wmma<<<64, 256, 0, stream>>>(bufA, cb, sg, pooled);
    // FC head -> (64,10)
    fc_head<<<64, 64, 0, stream>>>(pooled, l1w, l1b, l2w, l2b, outp);
}